// STDDAGMM_22041772163678
// MI455X (gfx1250) — compile-verified
//
#include <hip/hip_runtime.h>

#define BT 524288
#define DIN 118

typedef __attribute__((ext_vector_type(2))) float v2f;
typedef __attribute__((ext_vector_type(8))) float v8f;

// ---------------- weight prep: swizzle W[K][N] into zero-padded fragment order ----------
// wf[((t*NCH + c)*32 + lane)*2 + j] = W[k][n],  n = t*16 + (lane&15),
//                                     k = c*4 + (lane>>4)*2 + j   (0 outside K/N)
__global__ void dagmm_prep(const float* __restrict__ W, int K, int N,
                           float* __restrict__ wf) {
  int nchunks = (K + 3) >> 2;
  int ntiles  = (N + 15) >> 4;
  int total   = ntiles * nchunks * 64;
  int i = blockIdx.x * blockDim.x + threadIdx.x;
  if (i >= total) return;
  int j    = i & 1;
  int lane = (i >> 1) & 31;
  int tc   = i >> 6;
  int c    = tc % nchunks;
  int t    = tc / nchunks;
  int n = (t << 4) + (lane & 15);
  int k = (c << 2) + ((lane >> 4) << 1) + j;
  wf[i] = (k < K && n < N) ? W[k * N + n] : 0.0f;
}

// ---------------- one 16-row dense layer, branch-free WMMA inner loop ----------------
// ACT: 0=linear 1=relu 2=tanh.  Input LDS buffer must be zero-padded to ceil4(K) cols.
template<int K, int N, int ACT, bool GOUT>
__device__ __forceinline__ void dense16(const float* inb, int istr,
                                        float* outb, int ostr,
                                        const float* __restrict__ wf,
                                        const float* __restrict__ bias,
                                        int lane, float* gout) {
  constexpr int NCH = (K + 3) >> 2;
  constexpr int NT  = (N + 15) >> 4;
  constexpr int NP4 = (N + 3) & ~3;
  const float* arow = inb + (lane & 15) * istr + ((lane >> 4) << 1);
  #pragma unroll
  for (int t = 0; t < NT; ++t) {
    v8f acc = {0.f,0.f,0.f,0.f,0.f,0.f,0.f,0.f};
    const float* wt = wf + t * (NCH * 64) + (lane << 1);
    #pragma unroll
    for (int c = 0; c < NCH; ++c) {
      v2f a = *(const v2f*)(arow + (c << 2));   // ds_load_b64
      v2f b = *(const v2f*)(wt + (c << 6));     // global_load_b64, coalesced
      acc = __builtin_amdgcn_wmma_f32_16x16x4_f32(false, a, false, b, (short)0, acc,
                                                  false, false);
    }
    int n = (t << 4) + (lane & 15);
    if (n < NP4) {                              // epilogue only; zero-fills K-padding
      float bv = (n < N) ? bias[n] : 0.0f;
      int rbase = (lane >> 4) << 3;             // C-layout: reg r -> row r + 8*(lane>>4)
      #pragma unroll
      for (int r = 0; r < 8; ++r) {
        float v = acc[r] + bv;
        if (ACT == 1) v = fmaxf(v, 0.0f);
        if (ACT == 2) v = tanhf(v);
        if (n >= N) v = 0.0f;                   // pad column -> exact zero
        outb[(rbase + r) * ostr + n] = v;
        if (GOUT) { if (n < N) gout[(size_t)(rbase + r) * DIN + n] = v; }
      }
    }
  }
}

// ---------------- Pass A: MLP chain + z features + gamma + GMM sufficient stats --------
__global__ __launch_bounds__(64) void dagmm_passA(
    const float* __restrict__ x,
    const float* wf_e0, const float* eb0, const float* wf_e1, const float* eb1,
    const float* wf_e2, const float* eb2, const float* wf_e3, const float* eb3,
    const float* wf_d0, const float* db0, const float* wf_d1, const float* db1,
    const float* wf_d2, const float* db2, const float* wf_d3, const float* db3,
    const float* sW0, const float* sb0, const float* sW1, const float* sb1,
    float* __restrict__ dec_out, float* __restrict__ zws, float* __restrict__ bpart)
{
  __shared__ float xbuf[2][16 * 120];
  __shared__ float bufA[2][16 * 64];
  __shared__ float bufB[2][16 * 32];
  __shared__ float dbuf[2][16 * 120];
  __shared__ float encv[2][16];
  __shared__ float wsum[2][30];

  const int tid  = threadIdx.x;
  const int wave = tid >> 5;
  const int lane = tid & 31;
  const size_t rowbase = (((size_t)blockIdx.x << 1) + wave) << 4;

  float* xb = xbuf[wave];
  float* bA = bufA[wave];
  float* bB = bufB[wave];
  float* db = dbuf[wave];

  // stage x tile [16][118] (stride 120) and zero the 2 pad columns
  for (int i = lane; i < 16 * DIN; i += 32) {
    int r = i / DIN, cc = i - r * DIN;
    xb[r * 120 + cc] = x[(rowbase + r) * DIN + cc];
  }
  { int r = lane >> 1, cc = DIN + (lane & 1); xb[r * 120 + cc] = 0.0f; }
  __syncthreads();

  dense16<DIN,60,1,false>(xb, 120, bA, 64, wf_e0, eb0, lane, nullptr); __syncthreads();
  dense16< 60,30,1,false>(bA, 64,  bB, 32, wf_e1, eb1, lane, nullptr); __syncthreads();
  dense16< 30,10,1,false>(bB, 32,  bA, 64, wf_e2, eb2, lane, nullptr); __syncthreads();
  dense16< 10, 1,0,false>(bA, 64,  bB, 32, wf_e3, eb3, lane, nullptr); __syncthreads();
  if (lane < 16) encv[wave][lane] = bB[lane * 32];
  __syncthreads();
  dense16<  1,10,2,false>(bB, 32,  bA, 64, wf_d0, db0, lane, nullptr); __syncthreads();
  dense16< 10,30,2,false>(bA, 64,  bB, 32, wf_d1, db1, lane, nullptr); __syncthreads();
  dense16< 30,60,2,false>(bB, 32,  bA, 64, wf_d2, db2, lane, nullptr); __syncthreads();
  dense16< 60,DIN,0,true>(bA, 64,  db, 120, wf_d3, db3, lane, dec_out + rowbase * DIN);
  __syncthreads();

  // per-row reductions: 2 lanes per row, 59 columns each
  const int row  = lane & 15;
  const int half = lane >> 4;
  float sx = 0.f, sxx = 0.f, sdd = 0.f, sxd = 0.f, sdf = 0.f;
  const int c0 = half * 59;
  for (int c = c0; c < c0 + 59; ++c) {
    float xv = xb[row * 120 + c];
    float dv = db[row * 120 + c];
    sx += xv; sxx += xv * xv; sdd += dv * dv; sxd += xv * dv;
    float df = xv - dv; sdf += df * df;
  }
  sx  += __shfl_xor(sx, 16);
  sxx += __shfl_xor(sxx, 16);
  sdd += __shfl_xor(sdd, 16);
  sxd += __shfl_xor(sxd, 16);
  sdf += __shfl_xor(sdf, 16);

  float contrib[30];
  #pragma unroll
  for (int i = 0; i < 30; ++i) contrib[i] = 0.f;

  if (lane < 16) {
    float xn   = sqrtf(sxx);
    float dn   = sqrtf(sdd);
    float rel  = sqrtf(sdf) / fmaxf(xn, 1e-10f);
    float cs   = sxd / fmaxf(xn * dn, 1e-10f);
    float mean = sx * (1.0f / DIN);
    float var  = fmaxf(sxx * (1.0f / DIN) - mean * mean, 0.0f);
    float zv[4];
    zv[0] = encv[wave][row];
    zv[1] = rel; zv[2] = cs; zv[3] = sqrtf(var);
    size_t bidx = rowbase + row;
    zws[bidx * 4 + 0] = zv[0]; zws[bidx * 4 + 1] = zv[1];
    zws[bidx * 4 + 2] = zv[2]; zws[bidx * 4 + 3] = zv[3];

    // estimation net: tanh(z @ [4,10]) @ [10,2] -> softmax
    float l0 = sb1[0], l1 = sb1[1];
    #pragma unroll
    for (int j = 0; j < 10; ++j) {
      float h = sb0[j];
      #pragma unroll
      for (int g = 0; g < 4; ++g) h += sW0[g * 10 + j] * zv[g];
      h = tanhf(h);
      l0 += h * sW1[j * 2 + 0];
      l1 += h * sW1[j * 2 + 1];
    }
    float m = fmaxf(l0, l1);
    float e0 = expf(l0 - m), e1 = expf(l1 - m);
    float rinv = 1.0f / (e0 + e1);
    float g0 = e0 * rinv, g1 = e1 * rinv;

    // sufficient statistics: [0..1]=sum_g, [2..9]=sum_g*z, [10..29]=sum_g*z*z (upper tri)
    contrib[0] = g0; contrib[1] = g1;
    #pragma unroll
    for (int g = 0; g < 4; ++g) { contrib[2 + g] = g0 * zv[g]; contrib[6 + g] = g1 * zv[g]; }
    int idx = 10;
    #pragma unroll
    for (int g = 0; g < 4; ++g)
      for (int h2 = g; h2 < 4; ++h2) {
        float p = zv[g] * zv[h2];
        contrib[idx] = g0 * p; contrib[idx + 10] = g1 * p; ++idx;
      }
  }

  // deterministic in-wave tree reduction, then combine 2 waves
  #pragma unroll
  for (int i = 0; i < 30; ++i) {
    float v = contrib[i];
    v += __shfl_down(v, 16);
    v += __shfl_down(v, 8);
    v += __shfl_down(v, 4);
    v += __shfl_down(v, 2);
    v += __shfl_down(v, 1);
    if (lane == 0) wsum[wave][i] = v;
  }
  __syncthreads();
  if (tid < 30) bpart[(size_t)blockIdx.x * 30 + tid] = wsum[0][tid] + wsum[1][tid];
}

// ---------------- deterministic global reduce of block partials ----------------
__global__ void dagmm_reduce(const float* __restrict__ bpart, float* __restrict__ gacc,
                             int nblocks) {
  int t = threadIdx.x;
  if (t < 30) {
    float s = 0.f;
    for (int i = 0; i < nblocks; ++i) s += bpart[(size_t)i * 30 + t];
    gacc[t] = s;
  }
}

// ---------------- GMM parameter finalize ----------------
__device__ __forceinline__ void inv4(const float A[4][4], float inv[4][4], float* det) {
  float a[4][8];
  for (int i = 0; i < 4; ++i)
    for (int j = 0; j < 4; ++j) { a[i][j] = A[i][j]; a[i][j + 4] = (i == j) ? 1.f : 0.f; }
  float d = 1.f;
  for (int p = 0; p < 4; ++p) {
    int piv = p; float best = fabsf(a[p][p]);
    for (int r = p + 1; r < 4; ++r) { float v = fabsf(a[r][p]); if (v > best) { best = v; piv = r; } }
    if (piv != p) { for (int j = 0; j < 8; ++j) { float t = a[p][j]; a[p][j] = a[piv][j]; a[piv][j] = t; } d = -d; }
    float pv = a[p][p]; d *= pv;
    float ip = 1.0f / pv;
    for (int j = 0; j < 8; ++j) a[p][j] *= ip;
    for (int r = 0; r < 4; ++r) if (r != p) {
      float f = a[r][p];
      for (int j = 0; j < 8; ++j) a[r][j] -= f * a[p][j];
    }
  }
  for (int i = 0; i < 4; ++i) for (int j = 0; j < 4; ++j) inv[i][j] = a[i][j + 4];
  *det = d;
}

__global__ void dagmm_finalize(const float* __restrict__ gacc, const float* __restrict__ zws,
                               float* __restrict__ prm, float* __restrict__ cd_out) {
  if (threadIdx.x != 0) return;
  float sg[2] = {gacc[0], gacc[1]};
  float mu[2][4], rowsum[2][4], sqdet[2], phi[2];
  float covdiag = 0.f;
  for (int k = 0; k < 2; ++k) {
    phi[k] = sg[k] * (1.0f / (float)BT);
    for (int g = 0; g < 4; ++g) mu[k][g] = gacc[2 + k * 4 + g] / sg[k];
    float cov[4][4];
    int t = 0;
    for (int g = 0; g < 4; ++g)
      for (int h = g; h < 4; ++h) {
        float v = gacc[10 + k * 10 + t] / sg[k] - mu[k][g] * mu[k][h];
        cov[g][h] = v; cov[h][g] = v; ++t;
      }
    for (int g = 0; g < 4; ++g) cov[g][g] += 1e-12f;
    float ci[4][4]; float det;
    inv4(cov, ci, &det);
    sqdet[k] = sqrtf(det * 1558.5454565440383f);      // det(2*pi*cov) = (2pi)^4 det(cov)
    for (int g = 0; g < 4; ++g) covdiag += 1.0f / cov[g][g];
    for (int g = 0; g < 4; ++g) { float s = 0; for (int h = 0; h < 4; ++h) s += ci[g][h]; rowsum[k][g] = s; }
  }
  // reference quirk: scalar max_val taken from sample 0 only
  float z0[4] = {zws[0], zws[1], zws[2], zws[3]};
  float mv = 0.f;
  for (int k = 0; k < 2; ++k) {
    float s1 = 0.f, s2 = 0.f;
    for (int g = 0; g < 4; ++g) { float zm = z0[g] - mu[k][g]; s1 += zm * rowsum[k][g]; s2 += zm; }
    float et = -0.5f * s1 * s2;
    mv = fmaxf(mv, fmaxf(et, 0.f));
  }
  prm[0] = phi[0]; prm[1] = phi[1];
  for (int k = 0; k < 2; ++k)
    for (int g = 0; g < 4; ++g) { prm[2 + k * 4 + g] = mu[k][g]; prm[10 + k * 4 + g] = rowsum[k][g]; }
  prm[18] = sqdet[0]; prm[19] = sqdet[1]; prm[20] = mv;
  cd_out[0] = covdiag;
}

// ---------------- per-sample energy ----------------
__global__ __launch_bounds__(256) void dagmm_energy(const float* __restrict__ zws,
                                                    const float* __restrict__ prm,
                                                    float* __restrict__ en) {
  size_t b = (size_t)blockIdx.x * blockDim.x + threadIdx.x;
  float z[4] = {zws[b * 4], zws[b * 4 + 1], zws[b * 4 + 2], zws[b * 4 + 3]};
  float mv = prm[20];
  float s = 0.f;
  #pragma unroll
  for (int k = 0; k < 2; ++k) {
    float s1 = 0.f, s2 = 0.f;
    #pragma unroll
    for (int g = 0; g < 4; ++g) {
      float zm = z[g] - prm[2 + k * 4 + g];
      s1 += zm * prm[10 + k * 4 + g];
      s2 += zm;
    }
    float et = -0.5f * s1 * s2;
    float c = fminf(fmaxf(et - mv, -50.f), 50.f);
    s += prm[k] * expf(c) / (prm[18 + k] + 1e-12f);
  }
  en[b] = -mv - logf(s + 1e-12f);
}

// ---------------- launch ----------------
static inline int frag_elems(int K, int N) {
  return ((K + 3) >> 2) * ((N + 15) >> 4) * 64;
}

extern "C" void kernel_launch(void* const* d_in, const int* in_sizes, int n_in,
                              void* d_out, int out_size, void* d_ws, size_t ws_size,
                              hipStream_t stream) {
  (void)in_sizes; (void)n_in; (void)out_size; (void)ws_size;
  const float* x   = (const float*)d_in[0];
  const float* W[8]  = {(const float*)d_in[1], (const float*)d_in[3], (const float*)d_in[5],
                        (const float*)d_in[7], (const float*)d_in[9], (const float*)d_in[11],
                        (const float*)d_in[13], (const float*)d_in[15]};
  const float* bias[8] = {(const float*)d_in[2], (const float*)d_in[4], (const float*)d_in[6],
                          (const float*)d_in[8], (const float*)d_in[10], (const float*)d_in[12],
                          (const float*)d_in[14], (const float*)d_in[16]};
  const float* sW0 = (const float*)d_in[17]; const float* sb0 = (const float*)d_in[18];
  const float* sW1 = (const float*)d_in[19]; const float* sb1 = (const float*)d_in[20];

  const int Kd[8] = {DIN, 60, 30, 10, 1, 10, 30, 60};
  const int Nd[8] = {60, 30, 10, 1, 10, 30, 60, DIN};

  float* out     = (float*)d_out;
  float* dec_out = out;
  float* en_out  = out + (size_t)BT * DIN;
  float* cd_out  = en_out + BT;

  float* ws    = (float*)d_ws;
  float* gacc  = ws;                          // 32 floats
  float* prm   = ws + 32;                     // 32 floats
  float* zws   = ws + 64;                     // BT*4 floats
  const int nblocks = BT / 32;                // 2 waves x 16 rows per block
  float* bpart = ws + 64 + (size_t)BT * 4;    // nblocks*30 floats
  float* wbase = bpart + (size_t)nblocks * 30;

  // swizzle weights into fragment order (zero padded) -> branch-free WMMA loops
  float* wf[8];
  {
    float* p = wbase;
    for (int l = 0; l < 8; ++l) {
      wf[l] = p;
      int total = frag_elems(Kd[l], Nd[l]);
      dagmm_prep<<<(total + 255) / 256, 256, 0, stream>>>(W[l], Kd[l], Nd[l], wf[l]);
      p += total;
    }
  }

  dagmm_passA<<<nblocks, 64, 0, stream>>>(x,
      wf[0], bias[0], wf[1], bias[1], wf[2], bias[2], wf[3], bias[3],
      wf[4], bias[4], wf[5], bias[5], wf[6], bias[6], wf[7], bias[7],
      sW0, sb0, sW1, sb1, dec_out, zws, bpart);
  dagmm_reduce<<<1, 32, 0, stream>>>(bpart, gacc, nblocks);
  dagmm_finalize<<<1, 32, 0, stream>>>(gacc, zws, prm, cd_out);
  dagmm_energy<<<BT / 256, 256, 0, stream>>>(zws, prm, en_out);
}